// GMM_11321533792793
// MI455X (gfx1250) — compile-verified
//
#include <hip/hip_runtime.h>
#include <math.h>

#ifndef M_PI
#define M_PI 3.14159265358979323846
#endif

#define KCOMP 32
#define DDIM  128
#define NPTS  65536
#define EPS   1e-6f
#define MTILE 128       // points per block (8 waves x 16)
#define SASTRIDE 132    // padded LDS row stride (floats) -> conflict-free b64 reads

typedef float v2f __attribute__((ext_vector_type(2)));
typedef float v8f __attribute__((ext_vector_type(8)));
typedef unsigned int v4u __attribute__((ext_vector_type(4)));
typedef int v4i __attribute__((ext_vector_type(4)));
typedef int v8i __attribute__((ext_vector_type(8)));

// ---------------------------------------------------------------------------
// Kernel 1: per-component Cholesky of (cov + eps*I), A = L^{-1} (dense lower
// triangular, row-major [j][i]), b = A*mu, const_k = log w_k - sum log Ljj
// - (D/2) log 2pi.   grid = K blocks x 128 threads. Tiny cost vs main kernel.
// ---------------------------------------------------------------------------
__global__ void gmm_prep_kernel(const float* __restrict__ means,
                                const float* __restrict__ covs,
                                const float* __restrict__ weights,
                                float* __restrict__ wsA,   // [K][D][D]
                                float* __restrict__ wsB,   // [K][D]
                                float* __restrict__ wsC)   // [K]
{
    extern __shared__ float sm[];
    float* sM  = sm;                 // D*D
    float* sMu = sm + DDIM * DDIM;   // D
    const int k = blockIdx.x;
    const int t = threadIdx.x;

    for (int e = t; e < DDIM * DDIM; e += blockDim.x)
        sM[e] = covs[(size_t)k * DDIM * DDIM + e];
    if (t < DDIM) sMu[t] = means[k * DDIM + t];
    __syncthreads();
    if (t < DDIM) sM[t * DDIM + t] += EPS;
    __syncthreads();

    // In-place right-looking Cholesky; thread t owns row t (lower triangle).
    for (int j = 0; j < DDIM; ++j) {
        if (t == j) sM[j * DDIM + j] = sqrtf(sM[j * DDIM + j]);
        __syncthreads();
        float d = sM[j * DDIM + j];
        if (t > j) sM[t * DDIM + j] /= d;
        __syncthreads();
        if (t > j) {
            float lij = sM[t * DDIM + j];
            for (int c = j + 1; c <= t; ++c)
                sM[t * DDIM + c] -= lij * sM[c * DDIM + j];
        }
        __syncthreads();
    }

    if (t == 0) {
        float hl = 0.f;
        for (int j = 0; j < DDIM; ++j) hl += logf(sM[j * DDIM + j]);
        wsC[k] = logf(weights[k]) - hl - 0.5f * DDIM * logf(2.f * (float)M_PI);
    }

    // Forward substitution: thread t computes column t of A = L^{-1}.
    float a[DDIM];
    for (int i = 0; i < DDIM; ++i) {
        if (i < t) { a[i] = 0.f; continue; }
        float s = (i == t) ? 1.f : 0.f;
        for (int p = t; p < i; ++p) s -= sM[i * DDIM + p] * a[p];
        a[i] = s / sM[i * DDIM + i];
    }
    __syncthreads();   // everyone done reading L
    // Overwrite LDS with A row-major; also write A to workspace.
    for (int j = 0; j < DDIM; ++j) {
        sM[j * DDIM + t] = a[j];
        wsA[((size_t)k * DDIM + j) * DDIM + t] = a[j];
    }
    __syncthreads();
    // b[j] = sum_{i<=j} A[j][i] * mu[i]   (thread t computes row j = t)
    {
        float s = 0.f;
        for (int i = 0; i <= t; ++i) s += sM[t * DDIM + i] * sMu[i];
        wsB[k * DDIM + t] = s;
    }
}

// ---------------------------------------------------------------------------
// TDM: async-load one 128x128 f32 tile into LDS with 4-DWORD row padding
// (row stride becomes 132 floats -> matches SASTRIDE, conflict-free b64 reads)
// Descriptor per CDNA5 ISA (2D tensor, groups 2/3 unused).
// This toolchain exposes the 6-arg clang-23 builtin flavor.
// ---------------------------------------------------------------------------
__device__ __forceinline__ void tdm_load_A_tile(const float* gsrc,
                                                unsigned lds_byte_off) {
    unsigned long long ga = (unsigned long long)(uintptr_t)gsrc;
    v4u g0;
    g0.x = 1u;                                   // count=1, is_restore=0, no gather
    g0.y = lds_byte_off;                         // lds_addr (bytes)
    g0.z = (unsigned)(ga & 0xFFFFFFFFu);         // global_addr[31:0]
    g0.w = (unsigned)((ga >> 32) & 0x01FFFFFFu)  // global_addr[56:32]
           | (2u << 30);                         // type=2 ("image")
    v8i g1;
    // data_size=2 (4B) | pad_enable | pad_interval=6 (128 DW) | pad_amount=3 (4 DW)
    g1[0] = (int)((2u << 16) | (1u << 20) | (6u << 22) | (3u << 25));
    g1[1] = (int)(128u << 16);   // atomic_barrier_addr=0 | tensor_dim0[15:0]=128
    g1[2] = (int)(128u << 16);   // tensor_dim0[31:16]=0  | tensor_dim1[15:0]=128
    g1[3] = (int)(128u << 16);   // tensor_dim1[31:16]=0  | tile_dim0=128
    g1[4] = 128;                 // tile_dim1=128 | tile_dim2=0
    g1[5] = 128;                 // tensor_dim0_stride[31:0]=128
    g1[6] = (int)(16384u << 16); // dim0_stride[47:32]=0 | dim1_stride[15:0]=16384
    g1[7] = 0;                   // tensor_dim1_stride[47:16]=0
    v4i gz4 = {0, 0, 0, 0};
    v8i gz8 = {0, 0, 0, 0, 0, 0, 0, 0};
    __builtin_amdgcn_tensor_load_to_lds(g0, g1, gz4, gz4, gz8, 0);
}

// ---------------------------------------------------------------------------
// Kernel 2: main compute. Per block: 128 points, register-resident X
// fragments; loop k: TDM double-buffered A_k in LDS, triangular-skip f32
// WMMA GEMM, per-lane y^2 partials with ONE butterfly per k, online
// logsumexp in registers.
// ---------------------------------------------------------------------------
__global__ void __launch_bounds__(256)
gmm_main_kernel(const float* __restrict__ x,
                const float* __restrict__ wsA,
                const float* __restrict__ wsB,
                const float* __restrict__ wsC,
                float* __restrict__ out)
{
    extern __shared__ float sm[];
    float* sAbuf0 = sm;                       // DDIM x SASTRIDE floats
    float* sAbuf1 = sm + DDIM * SASTRIDE;     // DDIM x SASTRIDE floats

    const int tid  = threadIdx.x;
    const int lane = tid & 31;
    const int wave = tid >> 5;
    const int lrow = lane & 15;          // point row / output col within tile
    const int lhi  = lane >> 4;          // K-pair select (0: K0,K1  1: K2,K3)

    const int pbase = blockIdx.x * MTILE + wave * 16;

    // Register-resident X fragments in WMMA 16x4 f32 A-operand layout:
    // xf[c] = dims {4c+2*lhi, 4c+2*lhi+1} of point pbase+lrow.
    v2f xf[32];
    {
        const float* xp = x + (size_t)(pbase + lrow) * DDIM + 2 * lhi;
#pragma unroll
        for (int c = 0; c < 32; ++c)
            xf[c] = *(const v2f*)(xp + 4 * c);
    }

    float m8[8], s8[8];
#pragma unroll
    for (int r = 0; r < 8; ++r) { m8[r] = -INFINITY; s8[r] = 0.f; }

    // Prologue: async-load A_0 into buffer 0 (wave 0 issues, TDM ignores EXEC).
    if (wave == 0)
        tdm_load_A_tile(wsA, (unsigned)(uintptr_t)sAbuf0);

#pragma unroll 1
    for (int k = 0; k < KCOMP; ++k) {
        // Prefetch this component's b values (one per jt tile) so the loads
        // complete under the WMMA chains instead of stalling each jt.
        float bv[8];
#pragma unroll
        for (int jt = 0; jt < 8; ++jt)
            bv[jt] = wsB[k * DDIM + jt * 16 + lrow];
        const float kc = wsC[k];

        if (wave == 0)
            __builtin_amdgcn_s_wait_tensorcnt(0);   // A_k landed in LDS
        __syncthreads();                             // visible to all waves

        // Kick off A_{k+1} into the other buffer; overlaps with compute below.
        if (wave == 0 && (k + 1) < KCOMP)
            tdm_load_A_tile(wsA + (size_t)(k + 1) * DDIM * DDIM,
                            (unsigned)(uintptr_t)((k & 1) ? sAbuf0 : sAbuf1));

        const float* sA = (k & 1) ? sAbuf1 : sAbuf0;
        float maha[8];
#pragma unroll
        for (int r = 0; r < 8; ++r) maha[r] = 0.f;

#pragma unroll
        for (int jt = 0; jt < 8; ++jt) {        // output-dim tiles
            v8f acc = {};
#pragma unroll
            for (int it = 0; it < 8; ++it) {    // input-dim tiles
                if (it > jt) continue;          // A lower triangular: skip zeros
#pragma unroll
                for (int kk = 0; kk < 4; ++kk) {
                    const int i0 = it * 16 + kk * 4 + 2 * lhi;
                    v2f bf = *(const v2f*)(sA + (jt * 16 + lrow) * SASTRIDE + i0);
                    acc = __builtin_amdgcn_wmma_f32_16x16x4_f32(
                        false, xf[it * 4 + kk], false, bf,
                        (short)0, acc, false, false);
                }
            }
            // y = (A x)_j - b_j ; per-lane partial: this lane owns column
            // j = jt*16 + lrow, so accumulate y^2 locally (no shuffles here).
            const float bj = bv[jt];
#pragma unroll
            for (int r = 0; r < 8; ++r) {
                float y = acc[r] - bj;
                maha[r] = fmaf(y, y, maha[r]);
            }
        }

        // One 16-lane butterfly per k: sum partials over all 128 columns.
#pragma unroll
        for (int r = 0; r < 8; ++r) {
            float tsum = maha[r];
            tsum += __shfl_xor(tsum, 1);
            tsum += __shfl_xor(tsum, 2);
            tsum += __shfl_xor(tsum, 4);
            tsum += __shfl_xor(tsum, 8);
            // Online logsumexp over components (registers only).
            float v  = kc - 0.5f * tsum;
            float mn = fmaxf(m8[r], v);
            s8[r] = s8[r] * __expf(m8[r] - mn) + __expf(v - mn);
            m8[r] = mn;
        }
        __syncthreads();   // all waves done reading sA before it is re-filled
    }

    // D layout: VGPR r holds point r (lanes 0-15) / point r+8 (lanes 16-31).
    if (lane == 0) {
#pragma unroll
        for (int r = 0; r < 8; ++r) out[pbase + r] = m8[r] + __logf(s8[r]);
    } else if (lane == 16) {
#pragma unroll
        for (int r = 0; r < 8; ++r) out[pbase + 8 + r] = m8[r] + __logf(s8[r]);
    }
}

// ---------------------------------------------------------------------------
extern "C" void kernel_launch(void* const* d_in, const int* in_sizes, int n_in,
                              void* d_out, int out_size, void* d_ws, size_t ws_size,
                              hipStream_t stream) {
    const float* x       = (const float*)d_in[0];
    const float* means   = (const float*)d_in[1];
    const float* covs    = (const float*)d_in[2];
    const float* weights = (const float*)d_in[3];
    float* out = (float*)d_out;

    float* wsA = (float*)d_ws;                      // K*D*D floats (2 MB)
    float* wsB = wsA + (size_t)KCOMP * DDIM * DDIM; // K*D
    float* wsC = wsB + (size_t)KCOMP * DDIM;        // K

    const size_t lds1 = (DDIM * DDIM + DDIM) * sizeof(float);
    gmm_prep_kernel<<<KCOMP, DDIM, lds1, stream>>>(means, covs, weights,
                                                   wsA, wsB, wsC);

    const size_t lds2 = (size_t)2 * DDIM * SASTRIDE * sizeof(float);
    gmm_main_kernel<<<NPTS / MTILE, 256, lds2, stream>>>(x, wsA, wsB, wsC, out);
}